// HeteroGATLayer_61280593379564
// MI455X (gfx1250) — compile-verified
//
#include <hip/hip_runtime.h>
#include <hip/hip_bf16.h>

// ---------------------------------------------------------------------------
// HeteroGAT layer for MI455X (gfx1250, wave32).
// fp32 projections via V_WMMA_F32_16X16X4_F32; edge attention via wave-per-
// edge gather + shfl reduction + f32 global atomics; fused residual+ELU.
// ---------------------------------------------------------------------------

typedef __attribute__((ext_vector_type(2))) float v2f;
typedef __attribute__((ext_vector_type(8))) float v8f;

#define N_OP    200000
#define N_MAC   5000
#define IN_OP   64
#define IN_MAC  32
#define OUT_DIM 256
#define HEADS   4
#define DK      64      // OUT_DIM / HEADS
#define NEDGE   30000
#define EPSV    1e-6f

// ---- workspace layout (float offsets) -------------------------------------
#define WS_LIN_OP   ((size_t)0)
#define WS_LIN_MAC  (WS_LIN_OP  + (size_t)N_OP  * OUT_DIM)   // 51,200,000
#define WS_ACC_OP   (WS_LIN_MAC + (size_t)N_MAC * OUT_DIM)
#define WS_ACC_MAC  (WS_ACC_OP  + (size_t)N_OP  * OUT_DIM)
#define WS_DEN_SEQ  (WS_ACC_MAC + (size_t)N_MAC * OUT_DIM)
#define WS_DEN_OM   (WS_DEN_SEQ + (size_t)N_OP  * HEADS)
#define WS_DEN_MO   (WS_DEN_OM  + (size_t)N_MAC * HEADS)
#define WS_ALP_SEQ  (WS_DEN_MO  + (size_t)N_OP  * HEADS)
#define WS_ALP_OM   (WS_ALP_SEQ + (size_t)NEDGE * HEADS)
#define WS_ALP_MO   (WS_ALP_OM  + (size_t)NEDGE * HEADS)

// ---------------------------------------------------------------------------
// Zero-init kernel (accumulators + denominators region)
// ---------------------------------------------------------------------------
__global__ void zero_ws_kernel(float* __restrict__ p, size_t n) {
    size_t i = (size_t)blockIdx.x * blockDim.x + threadIdx.x;
    size_t stride = (size_t)gridDim.x * blockDim.x;
    for (; i < n; i += stride) p[i] = 0.0f;
}

// ---------------------------------------------------------------------------
// GEMM: out[M,256] = A[M,K] * W[256,K]^T + bias   (lin = h @ W.T + b)
// Block = 256 threads (8 waves). Block covers 16 rows of M; each wave does
// two 16-wide N-tiles (8 waves * 2 * 16 = 256 = OUT_DIM).
// Uses V_WMMA_F32_16X16X4_F32, K unrolled in steps of 4.
// ---------------------------------------------------------------------------
template <int K>
__global__ __launch_bounds__(256) void gemm_wmma_f32(
    const float* __restrict__ A, const float* __restrict__ W,
    const float* __restrict__ bias, float* __restrict__ out, int M) {
    const int lane = threadIdx.x & 31;
    const int wave = threadIdx.x >> 5;
    const int half = lane >> 4;     // 0: lanes 0-15, 1: lanes 16-31
    const int l16  = lane & 15;
    const int m_base = blockIdx.x * 16;

    // A fragment row (clamped so WMMA can run with full EXEC on ragged M)
    int arow = m_base + l16;
    if (arow >= M) arow = M - 1;
    const float* Ar = A + (size_t)arow * K;

    #pragma unroll
    for (int t = 0; t < 2; ++t) {
        const int n_base = (wave * 2 + t) * 16;
        const int ncol   = n_base + l16;
        const float* Wr  = W + (size_t)ncol * K;

        v8f c = {};
        #pragma unroll
        for (int k0 = 0; k0 < K; k0 += 4) {
            const int kk = k0 + 2 * half;     // per-lane K pair (ISA 7.12.2)
            v2f a, b;
            a.x = Ar[kk]; a.y = Ar[kk + 1];   // A 16x4: lane=M, vgpr=K
            b.x = Wr[kk]; b.y = Wr[kk + 1];   // B 4x16: lane=N, vgpr=K
            c = __builtin_amdgcn_wmma_f32_16x16x4_f32(
                    false, a, false, b, (short)0, c, false, false);
        }

        const float bn = bias[ncol];
        #pragma unroll
        for (int v = 0; v < 8; ++v) {
            const int mrow = m_base + v + 8 * half;   // C/D layout
            if (mrow < M)
                out[(size_t)mrow * OUT_DIM + ncol] = c[v] + bn;
        }
    }
}

// ---------------------------------------------------------------------------
// Edge scores: one wave32 per edge. Computes per-head LeakyReLU(score), exp,
// stores alpha[E,4] and atomically accumulates per-head denominators.
// att layout: [HEADS, 2*DK+1] = [4,129]
// ---------------------------------------------------------------------------
__global__ __launch_bounds__(256) void edge_score_kernel(
    const float* __restrict__ src_emb, const float* __restrict__ dst_emb,
    const int* __restrict__ src_idx, const int* __restrict__ dst_idx,
    const float* __restrict__ feat, const float* __restrict__ att,
    float* __restrict__ alpha, float* __restrict__ denom, int E) {
    const int e = blockIdx.x * 8 + (threadIdx.x >> 5);
    if (e >= E) return;
    const int lane = threadIdx.x & 31;
    const int s = src_idx[e];
    const int d = dst_idx[e];
    const float* hs = src_emb + (size_t)s * OUT_DIM;
    const float* hd = dst_emb + (size_t)d * OUT_DIM;

    float p[HEADS] = {0.f, 0.f, 0.f, 0.f};
    #pragma unroll
    for (int j = 0; j < 8; ++j) {
        const int dim = lane + 32 * j;     // head = j>>1 (uniform per step)
        const int h   = j >> 1;
        const int k   = dim & (DK - 1);
        p[h] += hs[dim] * att[h * (2 * DK + 1) + k]
              + hd[dim] * att[h * (2 * DK + 1) + DK + k];
    }
    // cross-lane reduction (wave32)
    #pragma unroll
    for (int off = 16; off > 0; off >>= 1) {
        p[0] += __shfl_xor(p[0], off, 32);
        p[1] += __shfl_xor(p[1], off, 32);
        p[2] += __shfl_xor(p[2], off, 32);
        p[3] += __shfl_xor(p[3], off, 32);
    }
    if (lane < HEADS) {
        const int h = lane;
        float sc = p[h] + feat[e] * att[h * (2 * DK + 1) + 2 * DK];
        sc = (sc >= 0.0f) ? sc : 0.2f * sc;              // LeakyReLU(0.2)
        const float a = expf(sc);
        alpha[(size_t)e * HEADS + h] = a;
        atomicAdd(&denom[(size_t)d * HEADS + h], a);
    }
}

// ---------------------------------------------------------------------------
// Edge scatter: one wave32 per edge, 8 dims/lane, f32 atomics into acc.
// ---------------------------------------------------------------------------
__global__ __launch_bounds__(256) void edge_scatter_kernel(
    const float* __restrict__ src_emb,
    const int* __restrict__ src_idx, const int* __restrict__ dst_idx,
    const float* __restrict__ alpha, const float* __restrict__ denom,
    float* __restrict__ acc, int E) {
    const int e = blockIdx.x * 8 + (threadIdx.x >> 5);
    if (e >= E) return;
    const int lane = threadIdx.x & 31;
    const int s = src_idx[e];
    const int d = dst_idx[e];
    const float* hs = src_emb + (size_t)s * OUT_DIM;

    float w[HEADS];
    #pragma unroll
    for (int h = 0; h < HEADS; ++h)
        w[h] = alpha[(size_t)e * HEADS + h] /
               (denom[(size_t)d * HEADS + h] + EPSV);

    float* out = acc + (size_t)d * OUT_DIM;
    #pragma unroll
    for (int j = 0; j < 8; ++j) {
        const int dim = lane + 32 * j;
        atomicAdd(&out[dim], hs[dim] * w[j >> 1]);
    }
}

// ---------------------------------------------------------------------------
// Residual + ELU: out = elu(acc + lin)
// ---------------------------------------------------------------------------
__global__ void residual_elu_kernel(const float* __restrict__ acc,
                                    const float* __restrict__ lin,
                                    float* __restrict__ out, size_t n) {
    size_t i = (size_t)blockIdx.x * blockDim.x + threadIdx.x;
    size_t stride = (size_t)gridDim.x * blockDim.x;
    for (; i < n; i += stride) {
        const float x = acc[i] + lin[i];
        out[i] = (x > 0.0f) ? x : (expf(x) - 1.0f);
    }
}

// ---------------------------------------------------------------------------
extern "C" void kernel_launch(void* const* d_in, const int* in_sizes, int n_in,
                              void* d_out, int out_size, void* d_ws, size_t ws_size,
                              hipStream_t stream) {
    const float* h_op     = (const float*)d_in[0];
    const float* h_mac    = (const float*)d_in[1];
    const int*   seq_src  = (const int*)d_in[2];
    const int*   seq_dst  = (const int*)d_in[3];
    const int*   om_src   = (const int*)d_in[4];
    const int*   om_dst   = (const int*)d_in[5];
    const int*   mo_src   = (const int*)d_in[6];
    const int*   mo_dst   = (const int*)d_in[7];
    const float* feat_seq = (const float*)d_in[8];
    const float* feat_om  = (const float*)d_in[9];
    const float* feat_mo  = (const float*)d_in[10];
    const float* W_op_w   = (const float*)d_in[11];
    const float* W_op_b   = (const float*)d_in[12];
    const float* W_mac_w  = (const float*)d_in[13];
    const float* W_mac_b  = (const float*)d_in[14];
    const float* att_seq  = (const float*)d_in[15];
    const float* att_om   = (const float*)d_in[16];
    const float* att_mo   = (const float*)d_in[17];

    float* ws = (float*)d_ws;
    float* lin_op  = ws + WS_LIN_OP;
    float* lin_mac = ws + WS_LIN_MAC;
    float* acc_op  = ws + WS_ACC_OP;
    float* acc_mac = ws + WS_ACC_MAC;
    float* den_seq = ws + WS_DEN_SEQ;
    float* den_om  = ws + WS_DEN_OM;
    float* den_mo  = ws + WS_DEN_MO;
    float* alp_seq = ws + WS_ALP_SEQ;
    float* alp_om  = ws + WS_ALP_OM;
    float* alp_mo  = ws + WS_ALP_MO;

    float* out_op  = (float*)d_out;
    float* out_mac = out_op + (size_t)N_OP * OUT_DIM;

    // 1) zero accumulators + denominators (contiguous region)
    {
        const size_t nz = WS_ALP_SEQ - WS_ACC_OP;
        zero_ws_kernel<<<8192, 256, 0, stream>>>(acc_op, nz);
    }

    // 2) projections via f32 WMMA
    gemm_wmma_f32<IN_OP><<<(N_OP + 15) / 16, 256, 0, stream>>>(
        h_op, W_op_w, W_op_b, lin_op, N_OP);
    gemm_wmma_f32<IN_MAC><<<(N_MAC + 15) / 16, 256, 0, stream>>>(
        h_mac, W_mac_w, W_mac_b, lin_mac, N_MAC);

    // 3) attention scores + denominators (one wave per edge)
    const int egrid = (NEDGE + 7) / 8;
    edge_score_kernel<<<egrid, 256, 0, stream>>>(
        lin_op, lin_op, seq_src, seq_dst, feat_seq, att_seq, alp_seq, den_seq, NEDGE);
    edge_score_kernel<<<egrid, 256, 0, stream>>>(
        lin_op, lin_mac, om_src, om_dst, feat_om, att_om, alp_om, den_om, NEDGE);
    edge_score_kernel<<<egrid, 256, 0, stream>>>(
        lin_mac, lin_op, mo_src, mo_dst, feat_mo, att_mo, alp_mo, den_mo, NEDGE);

    // 4) weighted scatter-adds
    edge_scatter_kernel<<<egrid, 256, 0, stream>>>(
        lin_op, seq_src, seq_dst, alp_seq, den_seq, acc_op, NEDGE);
    edge_scatter_kernel<<<egrid, 256, 0, stream>>>(
        lin_op, om_src, om_dst, alp_om, den_om, acc_mac, NEDGE);
    edge_scatter_kernel<<<egrid, 256, 0, stream>>>(
        lin_mac, mo_src, mo_dst, alp_mo, den_mo, acc_op, NEDGE);

    // 5) fused residual + ELU -> concatenated tuple output
    residual_elu_kernel<<<8192, 256, 0, stream>>>(
        acc_op, lin_op, out_op, (size_t)N_OP * OUT_DIM);
    residual_elu_kernel<<<1024, 256, 0, stream>>>(
        acc_mac, lin_mac, out_mac, (size_t)N_MAC * OUT_DIM);
}